// CustomRandModelForLrFinder_6210522710060
// MI455X (gfx1250) — compile-verified
//
#include <hip/hip_runtime.h>
#include <cstdint>
#include <cstddef>

// ---------------------------------------------------------------------------
// Fused anisotropic 25-tap bilinear sampling (B,1,512,512) for gfx1250.
// One thread = one output pixel. 16x16 pixel tile per 256-thread block; the
// input window (tile + 16px halo = 48x48 floats, 9.2KB LDS of 320KB/WGP) is
// staged via the CDNA5 async global->LDS path (global_load_async_to_lds_b32,
// ASYNCcnt-tracked). The LDS window pointer is laundered through an opaque
// asm operand so the optimizer cannot fold the ds_loads of the DMA-written
// window to undef (observed in earlier rounds). Rare samples whose offsets
// leave the window are re-gathered from global memory under a branch.
// ---------------------------------------------------------------------------

#define TILE 16
#define HALO 16
#define WIN  (TILE + 2 * HALO)      // 48
#define NTAP 5

#if __has_builtin(__builtin_amdgcn_global_load_async_to_lds_b32)
#define HAVE_ASYNC_LDS 1
#else
#define HAVE_ASYNC_LDS 0
#endif

typedef __attribute__((address_space(1))) int* g32_t;         // global int*
typedef __attribute__((address_space(3))) int* l32_t;         // LDS int*
typedef const __attribute__((address_space(3))) float* lds_cfp; // LDS const float*

__device__ __forceinline__ float frcp(float x) { return __builtin_amdgcn_rcpf(x); }

__global__ __launch_bounds__(256) void aniso_sample_kernel(
    const float* __restrict__ input,   // (B,1,H,W)
    const float* __restrict__ Lp,      // (B,H,W,3)
    const float* __restrict__ wch,     // (B,2,H,W)
    float* __restrict__ out)           // (B,1,H,W)
{
  constexpr int H = 512, W = 512;
  constexpr int tilesX = W / TILE;    // 32 (pow2 -> shifts, no sdiv)

  __shared__ float s_tile[WIN * WIN];

  const int blk = blockIdx.x;
  const int tX  = blk & (tilesX - 1);
  const int tY  = (blk >> 5) & (tilesX - 1);
  const int b   = blk >> 10;

  const int tid  = threadIdx.x;
  const int winY = tY * TILE - HALO;
  const int winX = tX * TILE - HALO;

  const float* img = input + (size_t)b * H * W;

  // ---- Stage 48x48 window (border-clamped) into LDS via async DMA ---------
  for (int i = tid; i < WIN * WIN; i += 256) {
    int ly = i / WIN, lx = i - ly * WIN;
    int gy = winY + ly; gy = gy < 0 ? 0 : (gy > H - 1 ? H - 1 : gy);
    int gx = winX + lx; gx = gx < 0 ? 0 : (gx > W - 1 ? W - 1 : gx);
#if HAVE_ASYNC_LDS
    __builtin_amdgcn_global_load_async_to_lds_b32(
        (g32_t)(uintptr_t)(img + (size_t)gy * W + gx),
        (l32_t)(uint32_t)(uintptr_t)&s_tile[i],
        0, 0);
#else
    s_tile[i] = img[(size_t)gy * W + gx];
#endif
  }
#if HAVE_ASYNC_LDS
#if __has_builtin(__builtin_amdgcn_s_wait_asynccnt)
  __builtin_amdgcn_s_wait_asynccnt(0);
#else
  asm volatile("s_wait_asynccnt 0x0" ::: "memory");
#endif
#endif
  __syncthreads();

  // Launder the LDS window base address through an opaque asm operand.
  // This (a) escapes s_tile so the optimizer must assume the async DMA wrote
  // it, and (b) rebuilds the pointer directly in addrspace(3), so the reads
  // below are guaranteed to lower to ds_load (never flat, never folded).
  uint32_t tbase = (uint32_t)(uintptr_t)&s_tile[0];
  asm volatile("" : "+s"(tbase) :: "memory");
  const lds_cfp tile = (lds_cfp)tbase;

  // ---- Per-pixel parameter math ------------------------------------------
  const int ty = tid >> 4, tx = tid & 15;
  const int y = tY * TILE + ty;
  const int x = tX * TILE + tx;

  const size_t pix = ((size_t)b * H + y) * W + x;
  const float* l = Lp + pix * 3;
  const float La = l[0], Lb = l[1], Lc = l[2];
  const size_t wbase = (size_t)b * 2 * H * W + (size_t)y * W + x;
  const float wx_ = wch[wbase];
  const float wy_ = wch[wbase + (size_t)H * W];

  const float a  = fabsf(La) + 1e-3f;
  const float bb = Lb;
  const float c  = fabsf(Lc) + 1e-3f;
  const float M00 = a * a;
  const float M01 = a * bb;
  const float M11 = bb * bb + c * c;
  const float d00 = M00 + 1e-6f;
  const float d11 = M11 + 1e-6f;
  const float det = d00 * d11 - M01 * M01;
  const float idt = frcp(det);
  const float I00 = d11 * idt, I01 = -M01 * idt, I11 = d00 * idt;

  const float q    = I00 * wx_ * wx_ + 2.f * I01 * wx_ * wy_ + I11 * wy_ * wy_;
  const float nrm  = sqrtf(q + 1e-6f);
  const float sig  = frcp(1.f + __expf(-nrm));
  const float tuned = (sig - 0.5f) * 2.f * (1.f - 1e-3f);
  const float scl  = tuned * frcp(nrm + 1e-6f);
  const float wtx = wx_ * scl, wty = wy_ * scl;

  // cos/sin of theta = 2*pi*j/5, j=0..4 (float32-rounded)
  const float CT[NTAP] = {1.f, 0.30901699f, -0.80901699f, -0.80901699f, 0.30901699f};
  const float ST[NTAP] = {0.f, 0.95105652f, 0.58778525f, -0.58778525f, -0.95105652f};
  const float SS[NTAP] = {0.2f, 0.4f, 0.6f, 0.8f, 1.0f};

  const float Hm1 = (float)(H - 1), Wm1 = (float)(W - 1);
  const float fwy0 = (float)winY, fwx0 = (float)winX;
  const float fys = (float)y, fxs = (float)x;

  float acc = 0.f;
#pragma unroll
  for (int j = 0; j < NTAP; ++j) {
    const float ct = CT[j], st = ST[j];
    const float quad = M00 * ct * ct + 2.f * M01 * ct * st + M11 * st * st;
    const float F = sqrtf(quad) + wtx * ct + wty * st;
    const float iF = frcp(F + 1e-6f);
    const float yx = ct * iF, yy = st * iF;
#pragma unroll
    for (int i = 0; i < NTAP; ++i) {
      const float s  = SS[i];
      const float py = fys + s * yy;
      const float px = fxs + s * yx;
      const float y0f = floorf(py), x0f = floorf(px);
      const float fy = py - y0f, fx = px - x0f;
      const float gy1 = 1.f - fy, gx1 = 1.f - fx;
      const float w00 = gy1 * gx1, w01 = gy1 * fx, w10 = fy * gx1, w11 = fy * fx;

      // Window-clamped LDS index: safe for any offset (NaN/huge -> 0).
      const float lyf = fminf(fmaxf(y0f - fwy0, 0.f), (float)(WIN - 2));
      const float lxf = fminf(fmaxf(x0f - fwx0, 0.f), (float)(WIN - 2));
      const int base = (int)lyf * WIN + (int)lxf;

      // Unconditional DS reads (common case).
      float v00 = tile[base];
      float v01 = tile[base + 1];
      float v10 = tile[base + WIN];
      float v11 = tile[base + WIN + 1];

      const bool inwin = (y0f >= fwy0) & (y0f <= fwy0 + (float)(WIN - 2)) &
                         (x0f >= fwx0) & (x0f <= fwx0 + (float)(WIN - 2));
      if (!inwin) {
        // Rare fallback: sample left the staged window; gather from global
        // with per-corner clamped indices (overwrites float values only).
        const float y0c = fminf(fmaxf(y0f,       0.f), Hm1);
        const float y1c = fminf(fmaxf(y0f + 1.f, 0.f), Hm1);
        const float x0c = fminf(fmaxf(x0f,       0.f), Wm1);
        const float x1c = fminf(fmaxf(x0f + 1.f, 0.f), Wm1);
        const int yi0 = (int)y0c, yi1 = (int)y1c;
        const int xi0 = (int)x0c, xi1 = (int)x1c;
        v00 = img[(size_t)yi0 * W + xi0];
        v01 = img[(size_t)yi0 * W + xi1];
        v10 = img[(size_t)yi1 * W + xi0];
        v11 = img[(size_t)yi1 * W + xi1];
      }

      // Corner validity exactly as the reference (mask applied after gather;
      // invalid corners contribute 0 regardless of the value read).
      const bool vy0 = (y0f >= 0.f)  & (y0f <= Hm1);
      const bool vy1 = (y0f >= -1.f) & (y0f <= Hm1 - 1.f);
      const bool vx0 = (x0f >= 0.f)  & (x0f <= Wm1);
      const bool vx1 = (x0f >= -1.f) & (x0f <= Wm1 - 1.f);
      acc += (vy0 & vx0) ? v00 * w00 : 0.f;
      acc += (vy0 & vx1) ? v01 * w01 : 0.f;
      acc += (vy1 & vx0) ? v10 * w10 : 0.f;
      acc += (vy1 & vx1) ? v11 * w11 : 0.f;
    }
  }

  out[pix] = acc * (1.f / 25.f);
}

extern "C" void kernel_launch(void* const* d_in, const int* in_sizes, int n_in,
                              void* d_out, int out_size, void* d_ws, size_t ws_size,
                              hipStream_t stream) {
  (void)n_in; (void)out_size; (void)d_ws; (void)ws_size;
  const float* input = (const float*)d_in[0];
  const float* Lp    = (const float*)d_in[1];
  const float* wch   = (const float*)d_in[2];
  float* out         = (float*)d_out;

  const int H = 512, W = 512;
  const int B = in_sizes[0] / (H * W);   // input is (B,1,H,W)

  const int blocks = B * (H / TILE) * (W / TILE);
  hipLaunchKernelGGL(aniso_sample_kernel, dim3(blocks), dim3(256), 0, stream,
                     input, Lp, wch, out);
}